// PhaseCoherenceLoss_69423851373142
// MI455X (gfx1250) — compile-verified
//
#include <hip/hip_runtime.h>

typedef __attribute__((ext_vector_type(2))) float v2f;
typedef __attribute__((ext_vector_type(8))) float v8f;

#define B_DIM 8
#define S_DIM 8192
#define C_DIM 256
#define NSC   16                  // s-chunks per (b, c-group)
#define S_SUB (S_DIM / NSC)       // 512 seq elements per wave
#define CG    16                  // channels per wave (M dimension)
#define NCG   (C_DIM / CG)        // 16 c-groups
#define NTILES (B_DIM * NCG * NSC)  // 2048 tiles == 2048 waves

// Phase 1: one wave per (b, c-group, s-chunk) tile.
// A-matrix (16x4 f32, documented layout): M = channel, K = seq.
//   lane L in [0,15]:  VGPR0 = (s+0, c0+L), VGPR1 = (s+1, c0+L)
//   lane L in [16,31]: VGPR0 = (s+2, c0+L-16), VGPR1 = (s+3, c0+L-16)
// B = all-ones 4x16 -> D[m][n] accumulates per-channel sums (replicated over n).
__global__ void PhaseCoh_phase1(const float* __restrict__ x,
                                float* __restrict__ wsum,
                                float* __restrict__ wsq) {
    const int tid  = blockIdx.x * blockDim.x + threadIdx.x;
    const int wave = tid >> 5;
    const int lane = tid & 31;

    const int b   = wave / (NCG * NSC);
    const int rem = wave % (NCG * NSC);
    const int cg  = rem / NSC;
    const int sc  = rem % NSC;
    const int c0  = cg * CG;
    const int s0  = sc * S_SUB;

    const int half = lane >> 4;     // 0: K={0,1}, 1: K={2,3}
    const int l    = lane & 15;     // channel within group

    // pointer at (b, s0 + 2*half, c0 + l); A.x at +0, A.y at +C_DIM
    const float* p = x + ((size_t)b * S_DIM + (size_t)(s0 + 2 * half)) * C_DIM
                       + c0 + l;

    v8f accS = {0.f, 0.f, 0.f, 0.f, 0.f, 0.f, 0.f, 0.f};
    v8f accQ = {0.f, 0.f, 0.f, 0.f, 0.f, 0.f, 0.f, 0.f};
    v2f ones; ones.x = 1.0f; ones.y = 1.0f;

    #pragma unroll 4
    for (int s = 0; s < S_SUB; s += 4) {
        const float ax = p[0];
        const float ay = p[C_DIM];
        v2f a; a.x = ax;      a.y = ay;
        v2f q; q.x = ax * ax; q.y = ay * ay;
        // D = A x ones + C : per-channel running sums in the matrix pipe
        accS = __builtin_amdgcn_wmma_f32_16x16x4_f32(
                   false, a, false, ones, (short)0, accS, false, false);
        accQ = __builtin_amdgcn_wmma_f32_16x16x4_f32(
                   false, q, false, ones, (short)0, accQ, false, false);
        p += 4 * C_DIM;
    }

    // C/D layout: VGPR r, lane 0  -> (M=r,   N=0); lane 16 -> (M=8+r, N=0).
    // So lanes 0 and 16 each hold 8 channel totals in accS[0..7]/accQ[0..7].
    if (l == 0) {
        const int baseIdx = wave * 16 + half * 8;
        #pragma unroll
        for (int r = 0; r < 8; ++r) {
            wsum[baseIdx + r] = accS[r];
            wsq [baseIdx + r] = accQ[r];
        }
    }
}

// Phase 2: fold the NSC partials per (b,c), compute variances, reduce to scalar.
__global__ void PhaseCoh_phase2(const float* __restrict__ wsum,
                                const float* __restrict__ wsq,
                                float* __restrict__ out) {
    __shared__ float red[256];
    const int t = threadIdx.x;

    float vacc = 0.f;
    #pragma unroll
    for (int k = 0; k < 8; ++k) {
        const int pair = k * 256 + t;          // b*256 + c
        const int b  = pair >> 8;
        const int c  = pair & 255;
        const int cg = c / CG;
        const int i  = c % CG;
        const int tbase = (b * NCG + cg) * NSC;
        float s = 0.f, q = 0.f;
        #pragma unroll
        for (int sc = 0; sc < NSC; ++sc) {
            const int idx = (tbase + sc) * 16 + i;
            s += wsum[idx];
            q += wsq[idx];
        }
        const float mean = s * (1.0f / (float)S_DIM);
        const float var  = q * (1.0f / (float)S_DIM) - mean * mean;
        vacc += var;
    }

    red[t] = vacc;
    __syncthreads();
    for (int off = 128; off > 0; off >>= 1) {
        if (t < off) red[t] += red[t + off];
        __syncthreads();
    }
    if (t == 0) {
        // 0.1 * 2 * mean_{b,c}(var)
        out[0] = 0.2f * red[0] * (1.0f / (float)(B_DIM * C_DIM));
    }
}

extern "C" void kernel_launch(void* const* d_in, const int* in_sizes, int n_in,
                              void* d_out, int out_size, void* d_ws, size_t ws_size,
                              hipStream_t stream) {
    (void)in_sizes; (void)n_in; (void)out_size; (void)ws_size;

    const float* phases = (const float*)d_in[0];
    float* out  = (float*)d_out;

    float* wsum = (float*)d_ws;                      // NTILES*16 floats
    float* wsq  = wsum + (size_t)NTILES * 16;        // NTILES*16 floats (256 KB total)

    // 2048 waves, 8 waves per 256-thread block -> 256 blocks
    const int threads = 256;
    const int blocks  = (NTILES * 32) / threads;     // 256

    PhaseCoh_phase1<<<blocks, threads, 0, stream>>>(phases, wsum, wsq);
    PhaseCoh_phase2<<<1, 256, 0, stream>>>(wsum, wsq, out);
}